// ResGatedGraphNet_64132451664028
// MI455X (gfx1250) — compile-verified
//
#include <hip/hip_runtime.h>
#include <hip/hip_bf16.h>
#include <math.h>

typedef __attribute__((ext_vector_type(2))) float v2f;
typedef __attribute__((ext_vector_type(8))) float v8f;

#define DF 64   // node feature dim
#define EDF 16  // edge feature dim

// ---------------------------------------------------------------------------
// Node GEMMs: k = h@Wk+bk, q = h@Wq+bq, v = h@Wv+bv, skip = h@Wskip+bias.
// One wave per (16-row tile, weight matrix). A strip (16x64 f32) hoisted into
// 32 VGPRs, then 4 N-tiles x 16 K-steps of V_WMMA_F32_16X16X4_F32.
// A layout (16x4): VGPR0 lanes0-15 K=0 / lanes16-31 K=2; VGPR1 K=1/K=3.
// B layout (4x16): VGPR0 lanes0-15 K=0 / lanes16-31 K=2; VGPR1 K=1/K=3.
// C/D (16x16): VGPR r -> rows r (lanes0-15) and r+8 (lanes16-31), col=lane&15.
// ---------------------------------------------------------------------------
__global__ void __launch_bounds__(256)
node_gemm4_kernel(const float* __restrict__ hin,
                  const float* __restrict__ Wk, const float* __restrict__ bk,
                  const float* __restrict__ Wq, const float* __restrict__ bq,
                  const float* __restrict__ Wv, const float* __restrict__ bv,
                  const float* __restrict__ Ws, const float* __restrict__ bs,
                  float* __restrict__ ko, float* __restrict__ qo,
                  float* __restrict__ vo, float* __restrict__ so,
                  int N)
{
    const int wave = (blockIdx.x * blockDim.x + threadIdx.x) >> 5;
    const int lane = threadIdx.x & 31;
    const int nWaves = (N >> 4) * 4;
    if (wave >= nWaves) return;

    const int m0    = (wave >> 2) << 4;  // 16-row tile base
    const int which = wave & 3;          // 0:k 1:q 2:v 3:skip

    const float* W; const float* b; float* out;
    switch (which) {
        case 0:  W = Wk; b = bk; out = ko; break;
        case 1:  W = Wq; b = bq; out = qo; break;
        case 2:  W = Wv; b = bv; out = vo; break;
        default: W = Ws; b = bs; out = so; break;
    }

    const int m    = lane & 15;
    const int half = lane >> 4;

    // Hoist A fragments for the full K=64 strip (16 steps of K=4).
    v2f afrag[16];
    const float* arow = hin + (size_t)(m0 + m) * DF + 2 * half;
    #pragma unroll
    for (int kk = 0; kk < 16; ++kk) {
        afrag[kk].x = arow[kk * 4 + 0];
        afrag[kk].y = arow[kk * 4 + 1];
    }

    #pragma unroll
    for (int nt = 0; nt < 4; ++nt) {
        const int n0 = nt << 4;
        const float bias = b[n0 + m];  // bias depends only on output column
        v8f c;
        #pragma unroll
        for (int r = 0; r < 8; ++r) c[r] = bias;

        #pragma unroll
        for (int kk = 0; kk < 16; ++kk) {
            const float* bp = W + (size_t)(kk * 4 + 2 * half) * DF + n0 + m;
            v2f bfrag;
            bfrag.x = bp[0];
            bfrag.y = bp[DF];
            c = __builtin_amdgcn_wmma_f32_16x16x4_f32(
                    false, afrag[kk], false, bfrag, (short)0, c, false, false);
        }

        #pragma unroll
        for (int r = 0; r < 8; ++r)
            out[(size_t)(m0 + r + half * 8) * DF + n0 + m] = c[r];
    }
}

// ---------------------------------------------------------------------------
// Edge phase, fully fused: e_d = be[d] + edge_attr[e,:] @ We[:,d] (We in LDS),
// gate = sigmoid(k[dst]+q[src]+e), msg = gate*v[src], atomic scatter-add into
// agg (which already holds h@Wskip+bias). 64 threads per edge (one per dim).
// ---------------------------------------------------------------------------
__global__ void __launch_bounds__(256)
edge_gate_scatter_kernel(const int* __restrict__ src, const int* __restrict__ dst,
                         const float* __restrict__ edge_attr,
                         const float* __restrict__ We, const float* __restrict__ be,
                         const float* __restrict__ kb, const float* __restrict__ qb,
                         const float* __restrict__ vb, float* __restrict__ agg,
                         int E)
{
    __shared__ float sWe[EDF * DF];
    __shared__ float sbe[DF];
    for (int i = threadIdx.x; i < EDF * DF; i += blockDim.x) sWe[i] = We[i];
    if (threadIdx.x < DF) sbe[threadIdx.x] = be[threadIdx.x];
    __syncthreads();

    const long long t = (long long)blockIdx.x * blockDim.x + threadIdx.x;
    const int e = (int)(t >> 6);
    const int d = (int)(t & 63);
    if (e >= E) return;

    const int s = src[e];
    const int i = dst[e];

    const float* ea = edge_attr + (size_t)e * EDF;
    float acc = sbe[d];
    #pragma unroll
    for (int j = 0; j < EDF; ++j) acc += ea[j] * sWe[j * DF + d];

    float g = kb[(size_t)i * DF + d] + qb[(size_t)s * DF + d] + acc;
    g = 1.0f / (1.0f + __expf(-g));
    float msg = g * vb[(size_t)s * DF + d];

    unsafeAtomicAdd(&agg[(size_t)i * DF + d], msg);  // global_atomic_add_f32
}

// ---------------------------------------------------------------------------
// In-place ELU (alpha = 1).
// ---------------------------------------------------------------------------
__global__ void __launch_bounds__(256)
elu_kernel(float* __restrict__ h, long long n)
{
    long long i = (long long)blockIdx.x * blockDim.x + threadIdx.x;
    if (i >= n) return;
    float x = h[i];
    h[i] = x > 0.0f ? x : (__expf(x) - 1.0f);
}

// ---------------------------------------------------------------------------
// Final head: logits = h3@Wl + bl; outputs:
//   out[0:N)              = sigmoid(logits)
//   out[N : N+N*64)       = feat (layer-2 activations)
//   out[N+N*64 : +N*65)   = concat(feat, logits) per row
// ---------------------------------------------------------------------------
__global__ void __launch_bounds__(256)
output_kernel(const float* __restrict__ h3, const float* __restrict__ feat,
              const float* __restrict__ Wl, const float* __restrict__ bl,
              float* __restrict__ out, int N)
{
    int i = blockIdx.x * blockDim.x + threadIdx.x;
    if (i >= N) return;

    const float* hr = h3 + (size_t)i * DF;
    float logit = bl[0];
    #pragma unroll
    for (int d = 0; d < DF; ++d) logit += hr[d] * Wl[d];

    out[i] = 1.0f / (1.0f + __expf(-logit));

    float* f_out = out + N + (size_t)i * DF;
    float* c_out = out + N + (size_t)N * DF + (size_t)i * (DF + 1);
    const float* fr = feat + (size_t)i * DF;
    #pragma unroll
    for (int d = 0; d < DF; ++d) {
        float fv = fr[d];
        f_out[d] = fv;
        c_out[d] = fv;
    }
    c_out[DF] = logit;
}

// ---------------------------------------------------------------------------
extern "C" void kernel_launch(void* const* d_in, const int* in_sizes, int n_in,
                              void* d_out, int out_size, void* d_ws, size_t ws_size,
                              hipStream_t stream)
{
    const float* x    = (const float*)d_in[0];
    const int*   ei   = (const int*)d_in[1];
    const float* ea   = (const float*)d_in[2];
    const float* Wk   = (const float*)d_in[3];
    const float* bk   = (const float*)d_in[4];
    const float* Wq   = (const float*)d_in[5];
    const float* bq   = (const float*)d_in[6];
    const float* Wv   = (const float*)d_in[7];
    const float* bv   = (const float*)d_in[8];
    const float* We   = (const float*)d_in[9];
    const float* be   = (const float*)d_in[10];
    const float* Wsk  = (const float*)d_in[11];
    const float* bias = (const float*)d_in[12];
    const float* Wl   = (const float*)d_in[13];
    const float* bl   = (const float*)d_in[14];

    const int N = in_sizes[0] / DF;
    const int E = in_sizes[1] / 2;
    const int* src = ei;       // edge_index[0]
    const int* dst = ei + E;   // edge_index[1]

    float* ws = (float*)d_ws;
    const size_t nd = (size_t)N * DF;
    float* kb = ws;
    float* qb = ws + nd;
    float* vb = ws + 2 * nd;
    float* h1 = ws + 3 * nd;
    float* h2 = ws + 4 * nd;  // = feat
    float* h3 = ws + 5 * nd;
    float* hbuf[3] = {h1, h2, h3};

    const int gemmWaves  = (N / 16) * 4;
    const int gemmBlocks = (gemmWaves * 32 + 255) / 256;
    const long long edgeThreads = (long long)E * DF;
    const int edgeBlocks = (int)((edgeThreads + 255) / 256);
    const int eluBlocks  = (int)((nd + 255) / 256);

    const float* hin = x;
    for (int l = 0; l < 3; ++l) {
        float* hout = hbuf[l];
        // k,q,v into scratch; skip+bias preloaded directly into hout (the
        // scatter accumulator), so no separate zero/add pass is needed.
        node_gemm4_kernel<<<gemmBlocks, 256, 0, stream>>>(
            hin,
            Wk  + (size_t)l * DF * DF, bk   + l * DF,
            Wq  + (size_t)l * DF * DF, bq   + l * DF,
            Wv  + (size_t)l * DF * DF, bv   + l * DF,
            Wsk + (size_t)l * DF * DF, bias + l * DF,
            kb, qb, vb, hout, N);

        edge_gate_scatter_kernel<<<edgeBlocks, 256, 0, stream>>>(
            src, dst, ea, We + (size_t)l * EDF * DF, be + l * DF,
            kb, qb, vb, hout, E);

        elu_kernel<<<eluBlocks, 256, 0, stream>>>(hout, (long long)nd);
        hin = hout;
    }

    output_kernel<<<(N + 255) / 256, 256, 0, stream>>>(
        h3, h2, Wl, bl, (float*)d_out, N);
}